// HyperGCN_Net_29188597744122
// MI455X (gfx1250) — compile-verified
//
#include <hip/hip_runtime.h>
#include <math.h>

// Problem constants (match reference)
#define NN   100000   // nodes
#define NE   50000    // hyperedges
#define NNZp 3200000  // incidences
#define FIN  512
#define HIDF 256      // 2^8
#define CLSF 16       // 2^4

typedef __attribute__((ext_vector_type(2))) float v2f;
typedef __attribute__((ext_vector_type(8))) float v8f;

// Only check the builtin on the DEVICE pass; the host (x86) pass of hipcc
// never defines amdgcn builtins and must not trip this.
#if defined(__HIP_DEVICE_COMPILE__) && !__has_builtin(__builtin_amdgcn_wmma_f32_16x16x4_f32)
#error "gfx1250 f32 WMMA builtin not available on this toolchain (device pass)"
#endif

static __device__ __forceinline__ v8f wmma_f32_k4(v2f a, v2f b, v8f c) {
  // D = A(16x4,f32) * B(4x16,f32) + C(16x16,f32)  -> v_wmma_f32_16x16x4_f32
  return __builtin_amdgcn_wmma_f32_16x16x4_f32(
      /*neg_a=*/false, a, /*neg_b=*/false, b,
      /*c_mod=*/(short)0, c, /*reuse_a=*/false, /*reuse_b=*/false);
}

// ---------------------------------------------------------------- utilities
__global__ void k_zero(float* __restrict__ p, long long n) {
  long long i = (long long)blockIdx.x * blockDim.x + threadIdx.x;
  long long s = (long long)gridDim.x * blockDim.x;
  for (; i < n; i += s) p[i] = 0.0f;
}

__global__ void k_deg(const int* __restrict__ nidx, const int* __restrict__ eidx,
                      float* __restrict__ dv, float* __restrict__ de) {
  int i = blockIdx.x * blockDim.x + threadIdx.x;
  if (i < NNZp) {
    atomicAdd(&dv[nidx[i]], 1.0f);
    atomicAdd(&de[eidx[i]], 1.0f);
  }
}

// dst[sidx[i]*F + f] += src[gidx[i]*F + f]   (F = 1<<LOGF)
// A wave covers consecutive f of the same incidence -> index loads are
// wave-uniform (F=256) and src reads / atomic dsts are lane-contiguous.
template <int LOGF>
__global__ void k_scatter(const float* __restrict__ src, const int* __restrict__ gidx,
                          const int* __restrict__ sidx, float* __restrict__ dst) {
  const int F = 1 << LOGF;
  const long long total = (long long)NNZp << LOGF;
  long long g = (long long)blockIdx.x * blockDim.x + threadIdx.x;
  long long stride = (long long)gridDim.x * blockDim.x;
  for (; g < total; g += stride) {
    long long i = g >> LOGF;
    int f = (int)(g & (F - 1));
    long long srow = gidx[i];
    long long drow = sidx[i];
    atomicAdd(dst + drow * F + f, src[srow * F + f]);
  }
}

template <int LOGF>
__global__ void k_scale_rows(float* __restrict__ feat, const float* __restrict__ deg,
                             long long total) {
  long long g = (long long)blockIdx.x * blockDim.x + threadIdx.x;
  long long stride = (long long)gridDim.x * blockDim.x;
  for (; g < total; g += stride) {
    float d = deg[g >> LOGF];
    feat[g] *= (d > 0.0f) ? (1.0f / d) : 0.0f;
  }
}

// out = relu(out * Dinv + b1), in place over [NN, 256]
__global__ void k_relu_bias(float* __restrict__ out, const float* __restrict__ dv,
                            const float* __restrict__ bias) {
  const long long total = (long long)NN * HIDF;
  long long g = (long long)blockIdx.x * blockDim.x + threadIdx.x;
  long long stride = (long long)gridDim.x * blockDim.x;
  for (; g < total; g += stride) {
    long long n = g >> 8;
    int f = (int)(g & 255);
    float d = dv[n];
    float inv = (d > 0.0f) ? (1.0f / d) : 0.0f;
    float v = out[g] * inv + bias[f];
    out[g] = fmaxf(v, 0.0f);
  }
}

// out = log_softmax(out * Dinv + b2) per row of 16, in place
__global__ void k_lsm(float* __restrict__ out, const float* __restrict__ dv,
                      const float* __restrict__ bias) {
  int n = blockIdx.x * blockDim.x + threadIdx.x;
  if (n >= NN) return;
  float d = dv[n];
  float inv = (d > 0.0f) ? (1.0f / d) : 0.0f;
  float v[CLSF];
  float m = -3.402823466e38f;
  long long base = (long long)n * CLSF;
#pragma unroll
  for (int c = 0; c < CLSF; ++c) {
    v[c] = out[base + c] * inv + bias[c];
    m = fmaxf(m, v[c]);
  }
  float s = 0.0f;
#pragma unroll
  for (int c = 0; c < CLSF; ++c) s += __expf(v[c] - m);
  float ls = __logf(s);
#pragma unroll
  for (int c = 0; c < CLSF; ++c) out[base + c] = v[c] - m - ls;
}

// ---------------------------------------------------------------- GEMM1
// C[100000,256] = A[100000,512] @ B[512,256], f32 WMMA 16x16x4.
// 8 waves/block: wave (mw in 0..1, nw in 0..3) -> 16(M) x 64(N) per wave.
__global__ __launch_bounds__(256) void k_gemm1(const float* __restrict__ A,
                                               const float* __restrict__ B,
                                               float* __restrict__ C) {
  const int K = FIN, N = HIDF;
  const int w = threadIdx.x >> 5;
  const int lane = threadIdx.x & 31;
  const int half = lane >> 4;   // 0: K pair {0,1}, 1: K pair {2,3}
  const int l = lane & 15;      // M (for A) / N (for B,C)
  const int m0 = blockIdx.x * 32 + (w >> 2) * 16;
  const int n0 = (w & 3) * 64;

  v8f acc0 = {}, acc1 = {}, acc2 = {}, acc3 = {};
  const float* arow = A + (size_t)(m0 + l) * K + 2 * half;
  for (int k0 = 0; k0 < K; k0 += 4) {
    v2f a;
    a.x = arow[k0];
    a.y = arow[k0 + 1];
    const float* bp = B + (size_t)(k0 + 2 * half) * N + n0 + l;
    v2f b;
    b.x = bp[0];       b.y = bp[N];        acc0 = wmma_f32_k4(a, b, acc0);
    b.x = bp[16];      b.y = bp[N + 16];   acc1 = wmma_f32_k4(a, b, acc1);
    b.x = bp[32];      b.y = bp[N + 32];   acc2 = wmma_f32_k4(a, b, acc2);
    b.x = bp[48];      b.y = bp[N + 48];   acc3 = wmma_f32_k4(a, b, acc3);
  }
  // C/D layout: VGPR r, lane l(+half*16) -> row m0 + r + 8*half, col n0 + l
  float* crow = C + (size_t)(m0 + 8 * half) * N + n0 + l;
#pragma unroll
  for (int r = 0; r < 8; ++r) {
    crow[(size_t)r * N + 0]  = acc0[r];
    crow[(size_t)r * N + 16] = acc1[r];
    crow[(size_t)r * N + 32] = acc2[r];
    crow[(size_t)r * N + 48] = acc3[r];
  }
}

// ---------------------------------------------------------------- GEMM2
// C[100000,16] = A[100000,256] @ B[256,16]; one 16x16 tile per wave.
__global__ __launch_bounds__(256) void k_gemm2(const float* __restrict__ A,
                                               const float* __restrict__ B,
                                               float* __restrict__ C) {
  const int K = HIDF, N = CLSF;
  const int w = threadIdx.x >> 5;
  const int lane = threadIdx.x & 31;
  const int half = lane >> 4;
  const int l = lane & 15;
  const int m0 = blockIdx.x * 128 + w * 16;
  if (m0 >= NN) return;  // whole-wave skip; EXEC stays all-1 for WMMA

  v8f acc = {};
  const float* arow = A + (size_t)(m0 + l) * K + 2 * half;
  const float* bcol = B + l;
  for (int k0 = 0; k0 < K; k0 += 4) {
    v2f a;
    a.x = arow[k0];
    a.y = arow[k0 + 1];
    int ka = k0 + 2 * half;
    v2f b;
    b.x = bcol[(size_t)ka * N];
    b.y = bcol[(size_t)(ka + 1) * N];
    acc = wmma_f32_k4(a, b, acc);
  }
  float* crow = C + (size_t)(m0 + 8 * half) * N + l;
#pragma unroll
  for (int r = 0; r < 8; ++r) crow[(size_t)r * N] = acc[r];
}

// ---------------------------------------------------------------- launch
extern "C" void kernel_launch(void* const* d_in, const int* in_sizes, int n_in,
                              void* d_out, int out_size, void* d_ws, size_t ws_size,
                              hipStream_t stream) {
  (void)in_sizes; (void)n_in; (void)out_size; (void)ws_size;

  const float* x  = (const float*)d_in[0];
  const int* hei  = (const int*)d_in[1];   // [2, NNZ]: row0 = node, row1 = edge
  const int* nidx = hei;
  const int* eidx = hei + NNZp;
  const float* W1 = (const float*)d_in[2];
  const float* b1 = (const float*)d_in[3];
  const float* W2 = (const float*)d_in[4];
  const float* b2 = (const float*)d_in[5];
  float* out = (float*)d_out;

  // Workspace layout (~257 MB): xw | e_feat | h | deg_v | deg_e
  float* ws    = (float*)d_ws;
  float* xw    = ws;                               // [NN, 256]  102.4 MB
  float* efeat = xw + (size_t)NN * HIDF;           // [NE, 256]   51.2 MB
  float* h     = efeat + (size_t)NE * HIDF;        // [NN, 256]  102.4 MB
  float* degv  = h + (size_t)NN * HIDF;            // [NN]
  float* dege  = degv + NN;                        // [NE]
  float* hw    = xw;     // reuse: xw dead after conv1 n->e scatter  [NN,16]
  float* ef2   = efeat;  // reuse: e_feat dead after conv1 e->n      [NE,16]

  const long long totH = (long long)NNZp << 8;   // 819.2M scatter elems (F=256)
  const long long totC = (long long)NNZp << 4;   // 51.2M  scatter elems (F=16)

  // degrees
  k_zero<<<1024, 256, 0, stream>>>(degv, (long long)(NN + NE));
  k_deg<<<(NNZp + 255) / 256, 256, 0, stream>>>(nidx, eidx, degv, dege);

  // layer 1: xw = x @ W1
  k_gemm1<<<NN / 32, 256, 0, stream>>>(x, W1, xw);

  // conv1 node->edge, scale by B^{-1}
  k_zero<<<8192, 256, 0, stream>>>(efeat, (long long)NE * HIDF);
  k_scatter<8><<<(int)((totH + 255) / 256), 256, 0, stream>>>(xw, nidx, eidx, efeat);
  k_scale_rows<8><<<(int)(((long long)NE * HIDF + 255) / 256), 256, 0, stream>>>(
      efeat, dege, (long long)NE * HIDF);

  // conv1 edge->node, scale by D^{-1}, +b1, relu (in place -> h)
  k_zero<<<8192, 256, 0, stream>>>(h, (long long)NN * HIDF);
  k_scatter<8><<<(int)((totH + 255) / 256), 256, 0, stream>>>(efeat, eidx, nidx, h);
  k_relu_bias<<<(int)(((long long)NN * HIDF + 255) / 256), 256, 0, stream>>>(h, degv, b1);

  // layer 2: hw = h @ W2
  k_gemm2<<<(NN / 16 + 7) / 8, 256, 0, stream>>>(h, W2, hw);

  // conv2 node->edge, B^{-1}
  k_zero<<<1024, 256, 0, stream>>>(ef2, (long long)NE * CLSF);
  k_scatter<4><<<(int)((totC + 255) / 256), 256, 0, stream>>>(hw, nidx, eidx, ef2);
  k_scale_rows<4><<<(int)(((long long)NE * CLSF + 255) / 256), 256, 0, stream>>>(
      ef2, dege, (long long)NE * CLSF);

  // conv2 edge->node into d_out, then fused D^{-1} + b2 + log_softmax
  k_zero<<<1024, 256, 0, stream>>>(out, (long long)NN * CLSF);
  k_scatter<4><<<(int)((totC + 255) / 256), 256, 0, stream>>>(ef2, eidx, nidx, out);
  k_lsm<<<(NN + 255) / 256, 256, 0, stream>>>(out, degv, b2);
}